// ALiBi_33981781246374
// MI455X (gfx1250) — compile-verified
//
#include <hip/hip_runtime.h>
#include <stdint.h>

// CDNA5 / gfx1250 — wave32, WMMA f32 16x16x4, async global->LDS staging.
typedef __attribute__((ext_vector_type(2))) float v2f;
typedef __attribute__((ext_vector_type(8))) float v8f;

#define SEQ 2048
#define HDIM 128
#define NHEAD 16
#define ROWPAD 132   // 128 + 4 floats: pitch mod 64 banks = 4 -> conflict-free column reads

// One 256-thread block (8 waves) computes a 128x128 tile of scores for one (b,h).
// q/k 128x128 tiles are staged into LDS once via async copies; WMMAs read LDS.
__global__ __launch_bounds__(256)
void alibi_wmma_kernel(const float* __restrict__ q,
                       const float* __restrict__ kmat,
                       const float* __restrict__ head_scales,
                       const float* __restrict__ slopes,
                       const float* __restrict__ positions,
                       const int*   __restrict__ token_indices,
                       float* __restrict__ out)
{
    __shared__ float qsh[128 * ROWPAD];
    __shared__ float ksh[128 * ROWPAD];

    const int bh   = blockIdx.z;          // b*NHEAD + h
    const int b    = bh >> 4;
    const int h    = bh & (NHEAD - 1);
    const int tid  = threadIdx.x;
    const int lane = tid & 31;
    const int wave = tid >> 5;            // 0..7
    const int half = lane >> 4;           // K-half for A/B frags, M-half for C/D
    const int l16  = lane & 15;

    const int m_blk  = blockIdx.y * 128;
    const int n_blk  = blockIdx.x * 128;
    const int m_base = m_blk + wave * 16;

    // ---- Async-stage the contiguous 64KB q and k tiles into padded LDS ----
    // Tile rows are contiguous in memory (full HDIM channels of 128 rows).
    const uint64_t qg = (uint64_t)(uintptr_t)(q    + ((size_t)bh * SEQ + m_blk) * HDIM);
    const uint64_t kg = (uint64_t)(uintptr_t)(kmat + ((size_t)bh * SEQ + n_blk) * HDIM);
    const uint32_t q_lds = (uint32_t)(uintptr_t)qsh;   // generic ptr low 32 bits = LDS offset
    const uint32_t k_lds = (uint32_t)(uintptr_t)ksh;

#pragma unroll
    for (int i = 0; i < 16; ++i) {
        const int c   = tid + i * 256;            // 16B-chunk id, 0..4095 (128 rows x 32 chunks)
        const int row = c >> 5;
        const int cir = c & 31;
        const uint32_t loff = (uint32_t)(row * (ROWPAD * 4) + cir * 16);
        const uint64_t goff = (uint64_t)c * 16;
        asm volatile("global_load_async_to_lds_b128 %0, %1, off"
                     :: "v"(q_lds + loff), "v"(qg + goff) : "memory");
        asm volatile("global_load_async_to_lds_b128 %0, %1, off"
                     :: "v"(k_lds + loff), "v"(kg + goff) : "memory");
    }
    asm volatile("s_wait_asynccnt 0x0" ::: "memory");
    __syncthreads();

    const float scale = head_scales[h] * 0.08838834764831845f; // 1/sqrt(128)
    const float slope = slopes[h];

    v8f acc[8];
#pragma unroll
    for (int t = 0; t < 8; ++t)
        acc[t] = (v8f){0.f, 0.f, 0.f, 0.f, 0.f, 0.f, 0.f, 0.f};

    // A frag: q row (wave*16 + l16), channels kb + 2*half + {0,1}
    const float* qrow = &qsh[(wave * 16 + l16) * ROWPAD + 2 * half];
    // B frag: k row (t*16 + l16), same channel pattern
    const float* krow = &ksh[l16 * ROWPAD + 2 * half];

#pragma unroll 4
    for (int kb = 0; kb < HDIM; kb += 4) {
        v2f a = *(const v2f*)(qrow + kb);
        a *= scale;
#pragma unroll
        for (int t = 0; t < 8; ++t) {
            v2f bf = *(const v2f*)(krow + t * 16 * ROWPAD + kb);
            // 8 args: (neg_a, A, neg_b, B, c_mod, C, reuse_a, reuse_b)
            acc[t] = __builtin_amdgcn_wmma_f32_16x16x4_f32(
                         false, a, false, bf, (short)0, acc[t], false, false);
        }
    }

    // ---- Epilogue: out[m][n] = acc - slope * (pos[m] - pos[n]) ----
    const int* tib = token_indices + (size_t)b * SEQ;

    float pq[8];
#pragma unroll
    for (int r = 0; r < 8; ++r)
        pq[r] = positions[tib[m_base + 8 * half + r]];

#pragma unroll
    for (int t = 0; t < 8; ++t) {
        const int n  = n_blk + t * 16 + l16;
        const float pk = positions[tib[n]];
#pragma unroll
        for (int r = 0; r < 8; ++r) {
            const int m = m_base + 8 * half + r;
            const float v = acc[t][r] - slope * (pq[r] - pk);
            // 537 MB streamed once: NT store keeps q/k resident in L2.
            __builtin_nontemporal_store(v, out + ((size_t)bh * SEQ + m) * SEQ + n);
        }
    }
}

extern "C" void kernel_launch(void* const* d_in, const int* in_sizes, int n_in,
                              void* d_out, int out_size, void* d_ws, size_t ws_size,
                              hipStream_t stream)
{
    const float* q             = (const float*)d_in[0];
    const float* k             = (const float*)d_in[1];
    const float* head_scales   = (const float*)d_in[2];
    const float* slopes        = (const float*)d_in[3];
    const float* positions     = (const float*)d_in[4];
    const int*   token_indices = (const int*)d_in[5];
    float* out = (float*)d_out;

    dim3 grid(SEQ / 128, SEQ / 128, 2 * NHEAD); // (16, 16, 32)
    dim3 block(256);
    alibi_wmma_kernel<<<grid, block, 0, stream>>>(q, k, head_scales, slopes,
                                                  positions, token_indices, out);
}